// DenseRMoK_81346680587023
// MI455X (gfx1250) — compile-verified
//
#include <hip/hip_runtime.h>
#include <math.h>
#include <stdint.h>

// ---------------- problem constants ----------------
#define BB 32
#define LL 512
#define NN 64
#define PP 96
#define EE 4
#define ROWS (BB * NN)          // 2048
#define MH_CONST 0.8673250705840776f   // 2/(sqrt(3)*pi^0.25)
#define REV_EPS 1e-5f
#define BN_EPS 1e-5f

typedef float v2f __attribute__((ext_vector_type(2)));
typedef float v8f __attribute__((ext_vector_type(8)));
typedef unsigned int v4u __attribute__((ext_vector_type(4)));
typedef int v4i __attribute__((ext_vector_type(4)));
typedef int v8i __attribute__((ext_vector_type(8)));

// ---------------- workspace layout (float offsets) ----------------
#define WS_XF     0                          // ROWS*LL            = 1048576
#define WS_MEAN   (WS_XF + ROWS * LL)        // ROWS               = 2048
#define WS_STD    (WS_MEAN + ROWS)           // ROWS
#define WS_SCORE  (WS_STD + ROWS)            // ROWS*4 (16B aligned)
#define WS_C1T    (WS_SCORE + ROWS * 4)      // 2*LL*PP            = 98304
#define WS_C2T    (WS_C1T + 2 * LL * PP)
#define WS_CONSTP (WS_C2T + 2 * LL * PP)     // 2*PP
#define WS_T      (WS_CONSTP + 2 * PP)       // 2*ROWS*PP
#define WS_W      (WS_T + 2 * ROWS * PP)     // 2*ROWS*PP

// =====================================================================
// TDM: DMA a 2D tile (tileW x tileH) out of a row-major fp32 tensor with
// row length rowLen into contiguous LDS at ldsByteOff.
// D# per CDNA5 ISA:
//  group0: [1:0]=count=1, [63:32]=lds_addr, [120:64]=global_addr,
//          [127:126]=type=2
//  group1: [17:16]=data_size(2 -> 4B), [79:48]=tensor_dim0,
//          [111:80]=tensor_dim1, [127:112]=tile_dim0, [143:128]=tile_dim1,
//          [207:160]=tensor_dim0_stride; tile_dim2=0 -> 2D, groups 2/3 zero.
// TDM writes X fastest, so tile rows of tileW floats lie back to back in LDS: the B-panel layout.
// =====================================================================
__device__ __forceinline__ void tdm_load_tile_2d(const void* gptr,
                                                 unsigned ldsByteOff,
                                                 unsigned rowLen,
                                                 unsigned tileW,
                                                 unsigned tileH) {
  const unsigned long long ga = (unsigned long long)(uintptr_t)gptr;
  v4u g0;
  g0[0] = 1u;                                    // count=1, user descriptor
  g0[1] = ldsByteOff;                            // lds_addr (bytes)
  g0[2] = (unsigned)ga;                          // global_addr[31:0]
  g0[3] = (unsigned)((ga >> 32) & 0x01ffffffu)   // global_addr[56:32]
          | (2u << 30);                          // type = 2 ("image")
  v8i g1;
  g1[0] = (int)(2u << 16);                       // data_size=4B; mask/pad=0
  g1[1] = (int)((rowLen & 0xffffu) << 16);       // tensor_dim0[15:0] @63:48
  g1[2] = (int)(((rowLen >> 16) & 0xffffu) |     // tensor_dim0[31:16]
                ((tileH & 0xffffu) << 16));      // tensor_dim1[15:0] @95:80
  g1[3] = (int)(((tileH >> 16) & 0xffffu) |      // tensor_dim1[31:16]
                ((tileW & 0xffffu) << 16));      // tile_dim0 @127:112
  g1[4] = (int)(tileH & 0xffffu);                // tile_dim1 @143:128
  g1[5] = (int)rowLen;                           // tensor_dim0_stride[31:0]
  g1[6] = 0;                                     // stride hi / dim1_stride lo
  g1[7] = 0;
  const v4i z4 = {0, 0, 0, 0};
#if __has_include(<hip/amd_detail/amd_gfx1250_TDM.h>)
  const v8i z8 = {0, 0, 0, 0, 0, 0, 0, 0};
  __builtin_amdgcn_tensor_load_to_lds(g0, g1, z4, z4, z8, 0);
#else
  __builtin_amdgcn_tensor_load_to_lds(g0, g1, z4, z4, 0);
#endif
}

// =====================================================================
// Kernel A: transpose taylor coeffs [P,L,3] -> C1t/C2t [e][L,P]
// =====================================================================
__global__ __launch_bounds__(PP) void prep_coeffs(
    const float* __restrict__ t0, const float* __restrict__ t1,
    float* __restrict__ C1t, float* __restrict__ C2t) {
  const int l = blockIdx.x, e = blockIdx.y, p = threadIdx.x;
  const float* C = e ? t1 : t0;
  const int src = (p * LL + l) * 3;
  const int dst = e * (LL * PP) + l * PP + p;
  C1t[dst] = C[src + 1];
  C2t[dst] = C[src + 2];
}

// Kernel B: constP[e][p] = sum_l coeffs[p,l,0] + bias[p]
__global__ __launch_bounds__(PP) void prep_const(
    const float* __restrict__ t0, const float* __restrict__ t1,
    const float* __restrict__ b0, const float* __restrict__ b1,
    float* __restrict__ constP) {
  const int e = blockIdx.x, p = threadIdx.x;
  const float* C = e ? t1 : t0;
  const float* bias = e ? b1 : b0;
  float s = 0.f;
  for (int l = 0; l < LL; ++l) s += C[(p * LL + l) * 3 + 0];
  constP[e * PP + p] = s + bias[p];
}

// =====================================================================
// Kernel 1: RevIN (mean/std over L) + normalize + gate softmax
// =====================================================================
__global__ __launch_bounds__(256) void revin_gate(
    const float* __restrict__ x, const float* __restrict__ rev_w,
    const float* __restrict__ rev_b, const float* __restrict__ gate_w,
    const float* __restrict__ gate_b, float* __restrict__ xf,
    float* __restrict__ meanArr, float* __restrict__ stdArr,
    float* __restrict__ score) {
  const int row = blockIdx.x;
  const int b = row >> 6, n = row & 63;
  const int tid = threadIdx.x;
  __shared__ float r1[256], r2[256];
  __shared__ float4 g4[256];

  const float* xb = x + (size_t)b * LL * NN + n;   // stride NN over l
  const float v0 = xb[(size_t)tid * NN];
  const float v1 = xb[(size_t)(tid + 256) * NN];
  r1[tid] = v0 + v1;
  r2[tid] = v0 * v0 + v1 * v1;
  __syncthreads();
  for (int s = 128; s > 0; s >>= 1) {
    if (tid < s) { r1[tid] += r1[tid + s]; r2[tid] += r2[tid + s]; }
    __syncthreads();
  }
  const float mean = r1[0] * (1.f / LL);
  const float var  = r2[0] * (1.f / LL) - mean * mean;
  const float stdev = sqrtf(var + REV_EPS);
  const float inv = 1.f / stdev;
  const float w = rev_w[n], bsh = rev_b[n];
  const float xn0 = (v0 - mean) * inv * w + bsh;
  const float xn1 = (v1 - mean) * inv * w + bsh;
  xf[(size_t)row * LL + tid] = xn0;
  xf[(size_t)row * LL + tid + 256] = xn1;

  float4 g;
  g.x = xn0 * gate_w[0 * LL + tid] + xn1 * gate_w[0 * LL + tid + 256];
  g.y = xn0 * gate_w[1 * LL + tid] + xn1 * gate_w[1 * LL + tid + 256];
  g.z = xn0 * gate_w[2 * LL + tid] + xn1 * gate_w[2 * LL + tid + 256];
  g.w = xn0 * gate_w[3 * LL + tid] + xn1 * gate_w[3 * LL + tid + 256];
  g4[tid] = g;
  __syncthreads();
  for (int s = 128; s > 0; s >>= 1) {
    if (tid < s) {
      float4 a = g4[tid], c = g4[tid + s];
      a.x += c.x; a.y += c.y; a.z += c.z; a.w += c.w;
      g4[tid] = a;
    }
    __syncthreads();
  }
  if (tid == 0) {
    float lg[EE];
    lg[0] = g4[0].x + gate_b[0]; lg[1] = g4[0].y + gate_b[1];
    lg[2] = g4[0].z + gate_b[2]; lg[3] = g4[0].w + gate_b[3];
    float mx = lg[0];
    for (int e = 1; e < EE; ++e) mx = fmaxf(mx, lg[e]);
    float sum = 0.f;
    for (int e = 0; e < EE; ++e) { lg[e] = expf(lg[e] - mx); sum += lg[e]; }
    const float isum = 1.f / sum;
    for (int e = 0; e < EE; ++e) score[row * 4 + e] = lg[e] * isum;
    meanArr[row] = mean;
    stdArr[row] = stdev;
  }
}

// =====================================================================
// Kernel 2: Taylor experts, fp32 WMMA GEMM with TDM-staged B panels.
//   T[e] = xf @ C1[e] + xf^2 @ C2[e] + constP[e]
// grid (ROWS/128, PP/16, 2), 256 thr = 8 waves; block owns 8 M-tiles x
// one 16-wide N tile. Wave 0 DMAs both 512x16 B panels (C1,C2) into LDS
// via tensor_load_to_lds, waits on TENSORcnt, then all 8 waves run the
// K-loop reading B fragments from LDS (conflict-free: 16 consecutive
// banks per half-wave).
// A fragment (16x4 f32, ISA 7.12.2): lane m=lane&15, half=lane>>4 holds
//   {x[m][k0+2*half], x[m][k0+2*half+1]}; B striped symmetrically.
// =====================================================================
__global__ __launch_bounds__(256) void taylor_wmma(
    const float* __restrict__ xf, const float* __restrict__ C1t,
    const float* __restrict__ C2t, const float* __restrict__ constP,
    float* __restrict__ T) {
  __shared__ float Bs1[LL * 16];   // 32 KB
  __shared__ float Bs2[LL * 16];   // 32 KB
  const int nTile = blockIdx.y, e = blockIdx.z;
  const int tid = threadIdx.x;
  const int wave = tid >> 5, lane = tid & 31;
  const int half = lane >> 4;
  const int mRow = lane & 15;
  const int nCol = lane & 15;
  const int p0 = nTile * 16;

  if (tid < 32) {   // wave 0 issues the two tensor DMAs
    tdm_load_tile_2d(C1t + (size_t)e * LL * PP + p0,
                     (unsigned)(uintptr_t)Bs1, PP, 16, LL);
    tdm_load_tile_2d(C2t + (size_t)e * LL * PP + p0,
                     (unsigned)(uintptr_t)Bs2, PP, 16, LL);
    __builtin_amdgcn_s_wait_tensorcnt(0);
  }
  __syncthreads();

  const int mTile = blockIdx.x * 8 + wave;
  const float* __restrict__ Arow =
      xf + (size_t)(mTile * 16 + mRow) * LL + 2 * half;

  v8f acc = {};
#pragma unroll 4
  for (int k0 = 0; k0 < LL; k0 += 4) {
    const v2f a = *(const v2f*)(Arow + k0);   // aligned b64 load
    const int kb = k0 + 2 * half;
    v2f b1, b2;
    b1.x = Bs1[kb * 16 + nCol];
    b1.y = Bs1[(kb + 1) * 16 + nCol];
    b2.x = Bs2[kb * 16 + nCol];
    b2.y = Bs2[(kb + 1) * 16 + nCol];
    __builtin_prefetch(Arow + k0 + 32, 0, 0);
    // x channel
    acc = __builtin_amdgcn_wmma_f32_16x16x4_f32(
        false, a, false, b1, (short)0, acc, false, false);
    // x^2 channel: lanewise square of A fragment == fragment of x^2
    v2f a2; a2.x = a.x * a.x; a2.y = a.y * a.y;
    acc = __builtin_amdgcn_wmma_f32_16x16x4_f32(
        false, a2, false, b2, (short)0, acc, false, false);
  }

  const float cp = constP[e * PP + p0 + nCol];
  float* __restrict__ Te = T + (size_t)e * ROWS * PP;
#pragma unroll
  for (int g = 0; g < 8; ++g) {
    const int row = mTile * 16 + 8 * half + g;   // C/D layout: M = g + 8*half
    Te[(size_t)row * PP + p0 + nCol] = acc[g] + cp;
  }
}

// =====================================================================
// Kernel 3: WaveKAN (mexican hat) experts. Block (8 waves) per row;
// each wave owns (e,p) tasks, lanes stride over l, __shfl_xor reduce.
// =====================================================================
__global__ __launch_bounds__(256) void wave_experts(
    const float* __restrict__ xf,
    const float* __restrict__ s0, const float* __restrict__ tr0,
    const float* __restrict__ ww0, const float* __restrict__ ga0,
    const float* __restrict__ be0,
    const float* __restrict__ s1, const float* __restrict__ tr1,
    const float* __restrict__ ww1, const float* __restrict__ ga1,
    const float* __restrict__ be1,
    float* __restrict__ W) {
  const int row = blockIdx.x;
  const int tid = threadIdx.x;
  __shared__ float lx[LL];
  lx[tid] = xf[(size_t)row * LL + tid];
  lx[tid + 256] = xf[(size_t)row * LL + tid + 256];
  __syncthreads();

  const int wave = tid >> 5, lane = tid & 31;
  const float bnScale = 1.0f / sqrtf(1.0f + BN_EPS);

  for (int task = wave; task < 2 * PP; task += 8) {
    const int e = task >= PP;
    const int p = task - e * PP;
    const float* __restrict__ sc = e ? s1 : s0;
    const float* __restrict__ tr = e ? tr1 : tr0;
    const float* __restrict__ wv = e ? ww1 : ww0;
    const int base = p * LL;
    float acc = 0.f;
#pragma unroll 4
    for (int l = lane; l < LL; l += 32) {
      const float xs = (lx[l] - tr[base + l]) / sc[base + l];
      const float x2 = xs * xs;
      const float psi = MH_CONST * (x2 - 1.f) * expf(-0.5f * x2);
      acc += psi * wv[base + l];
    }
    for (int off = 16; off > 0; off >>= 1) acc += __shfl_xor(acc, off, 32);
    if (lane == 0) {
      const float gamma = e ? ga1[p] : ga0[p];
      const float beta  = e ? be1[p] : be0[p];
      W[(size_t)e * ROWS * PP + (size_t)row * PP + p] =
          acc * bnScale * gamma + beta;
    }
  }
}

// =====================================================================
// Kernel 4: gate-weighted mixture + RevIN denorm -> out [B,P,N]
// =====================================================================
__global__ __launch_bounds__(PP) void combine(
    const float* __restrict__ T, const float* __restrict__ W,
    const float* __restrict__ score, const float* __restrict__ meanArr,
    const float* __restrict__ stdArr, const float* __restrict__ rev_w,
    const float* __restrict__ rev_b, float* __restrict__ out) {
  const int row = blockIdx.x;
  const int b = row >> 6, n = row & 63;
  const int p = threadIdx.x;
  const float4 s = *(const float4*)(score + row * 4);
  const size_t idx = (size_t)row * PP + p;
  const float pred = T[idx] * s.x +
                     T[(size_t)ROWS * PP + idx] * s.y +
                     W[idx] * s.z +
                     W[(size_t)ROWS * PP + idx] * s.w;
  const float o = ((pred - rev_b[n]) / (rev_w[n] + REV_EPS)) * stdArr[row] +
                  meanArr[row];
  out[((size_t)b * PP + p) * NN + n] = o;
}

// =====================================================================
extern "C" void kernel_launch(void* const* d_in, const int* in_sizes, int n_in,
                              void* d_out, int out_size, void* d_ws,
                              size_t ws_size, hipStream_t stream) {
  (void)in_sizes; (void)n_in; (void)out_size; (void)ws_size;
  const float* x      = (const float*)d_in[0];
  const float* rev_w  = (const float*)d_in[1];
  const float* rev_b  = (const float*)d_in[2];
  const float* gate_w = (const float*)d_in[3];
  const float* gate_b = (const float*)d_in[4];
  const float* t0c    = (const float*)d_in[5];
  const float* t0b    = (const float*)d_in[6];
  const float* t1c    = (const float*)d_in[7];
  const float* t1b    = (const float*)d_in[8];
  const float* w0s    = (const float*)d_in[9];
  const float* w0t    = (const float*)d_in[10];
  const float* w0w    = (const float*)d_in[11];
  const float* w0g    = (const float*)d_in[12];
  const float* w0be   = (const float*)d_in[13];
  const float* w1s    = (const float*)d_in[14];
  const float* w1t    = (const float*)d_in[15];
  const float* w1w    = (const float*)d_in[16];
  const float* w1g    = (const float*)d_in[17];
  const float* w1be   = (const float*)d_in[18];

  float* ws = (float*)d_ws;
  float* xf     = ws + WS_XF;
  float* meanA  = ws + WS_MEAN;
  float* stdA   = ws + WS_STD;
  float* score  = ws + WS_SCORE;
  float* C1t    = ws + WS_C1T;
  float* C2t    = ws + WS_C2T;
  float* constP = ws + WS_CONSTP;
  float* T      = ws + WS_T;
  float* W      = ws + WS_W;

  prep_coeffs<<<dim3(LL, 2), PP, 0, stream>>>(t0c, t1c, C1t, C2t);
  prep_const<<<2, PP, 0, stream>>>(t0c, t1c, t0b, t1b, constP);
  revin_gate<<<ROWS, 256, 0, stream>>>(x, rev_w, rev_b, gate_w, gate_b, xf,
                                       meanA, stdA, score);
  taylor_wmma<<<dim3(ROWS / 128, PP / 16, 2), 256, 0, stream>>>(xf, C1t, C2t,
                                                                constP, T);
  wave_experts<<<ROWS, 256, 0, stream>>>(xf, w0s, w0t, w0w, w0g, w0be, w1s,
                                         w1t, w1w, w1g, w1be, W);
  combine<<<ROWS, PP, 0, stream>>>(T, W, score, meanA, stdA, rev_w, rev_b,
                                   (float*)d_out);
}